// QiskitQuantumLayer_16097537425539
// MI455X (gfx1250) — compile-verified
//
#include <hip/hip_runtime.h>
#include <hip/hip_bf16.h>

#define NQ       14
#define NSTATES  (1 << NQ)          // 16384 amplitudes
#define NLAYERS  4
#define THREADS  256                // 8 wave32s
#define NWAVES   (THREADS / 32)
#define NPAIRS   (NSTATES / 2)      // 8192 pairs per single-qubit gate
#define NTILES   (NPAIRS / 16)      // 512 WMMA tiles per gate pass
#define TPW      (NTILES / NWAVES)  // 64 tiles per wave per pass

typedef __attribute__((ext_vector_type(2))) float v2f;   // 16x4 f32 A/B operand = 2 VGPRs
typedef __attribute__((ext_vector_type(8))) float v8f;   // 16x16 f32 C/D operand = 8 VGPRs

__device__ __forceinline__ float2 cmul(float2 a, float2 b) {
  return make_float2(__fmaf_rn(a.x, b.x, -a.y * b.y),
                     __fmaf_rn(a.x, b.y,  a.y * b.x));
}
__device__ __forceinline__ float2 cadd(float2 a, float2 b) {
  return make_float2(a.x + b.x, a.y + b.y);
}
__device__ __forceinline__ float2 cscale(float s, float2 a) {
  return make_float2(s * a.x, s * a.y);
}

// Fused single-qubit gate G = RX(c) * RZ(b) * RY(a)   (circuit applies RY,RZ,RX in order)
__device__ __forceinline__ void fuse_gate(float a, float b, float c,
                                          float2& g00, float2& g01,
                                          float2& g10, float2& g11) {
  float cy = cosf(0.5f * a), sy = sinf(0.5f * a);
  float cz = cosf(0.5f * b), sz = sinf(0.5f * b);
  float cx = cosf(0.5f * c), sx = sinf(0.5f * c);
  float2 em = make_float2(cz, -sz);          // e^{-i b/2}
  float2 ep = make_float2(cz,  sz);          // e^{+i b/2}
  float2 m00 = cscale(cy, em), m01 = cscale(-sy, em);   // RZ*RY
  float2 m10 = cscale(sy, ep), m11 = cscale( cy, ep);
  float2 is  = make_float2(0.f, -sx);        // -i sin(c/2)
  g00 = cadd(cscale(cx, m00), cmul(is, m10));
  g01 = cadd(cscale(cx, m01), cmul(is, m11));
  g10 = cadd(cmul(is, m00), cscale(cx, m10));
  g11 = cadd(cmul(is, m01), cscale(cx, m11));
}

__global__ __launch_bounds__(THREADS)
void qsim14_kernel(const float* __restrict__ x,      // (B, 14)
                   const float* __restrict__ w,      // (4, 14, 3)
                   float* __restrict__ out) {        // (B, 14)
  extern __shared__ float smem[];
  float2* bufA = (float2*)smem;                 // 16384 complex (128 KB)
  float2* bufB = bufA + NSTATES;                // 16384 complex (128 KB) perm double-buffer
  float2* gm   = bufB + NSTATES;                // 14 gates * 4 complex
  float2* vin  = gm + NQ * 4;                   // 14 qubits * 2 complex
  float*  red  = (float*)(vin + NQ * 2);        // 14 reduction slots

  const int tid  = threadIdx.x;
  const int lane = tid & 31;                    // wave32
  const int wid  = tid >> 5;                    // 8 waves
  const int b    = blockIdx.x;

  if (tid < NQ) red[tid] = 0.f;
  if (tid == 0) __builtin_prefetch(&x[b * NQ], 0, 0);   // global_prefetch for angles

  // ---- Product-state init: data RY(x) fused with ALL layer-0 rotations ----
  if (tid < NQ) {
    float2 g00, g01, g10, g11;
    fuse_gate(w[(0 * NQ + tid) * 3 + 0], w[(0 * NQ + tid) * 3 + 1],
              w[(0 * NQ + tid) * 3 + 2], g00, g01, g10, g11);
    float h  = 0.5f * x[b * NQ + tid];
    float ct = cosf(h), st = sinf(h);           // RY(x)|0> = (cos, sin)
    vin[tid * 2 + 0] = cadd(cscale(ct, g00), cscale(st, g01));
    vin[tid * 2 + 1] = cadd(cscale(ct, g10), cscale(st, g11));
  }
  __syncthreads();

  // product tables: low 8 qubits -> bufB[0..255], high 6 qubits -> bufB[256..319]
  {
    float2 r = vin[0 * 2 + (tid & 1)];
    #pragma unroll
    for (int j = 1; j < 8; ++j) r = cmul(r, vin[j * 2 + ((tid >> j) & 1)]);
    bufB[tid] = r;
    if (tid < 64) {
      float2 h2 = vin[8 * 2 + (tid & 1)];
      #pragma unroll
      for (int j = 1; j < 6; ++j) h2 = cmul(h2, vin[(8 + j) * 2 + ((tid >> j) & 1)]);
      bufB[256 + tid] = h2;
    }
  }
  __syncthreads();
  for (int m = tid; m < NSTATES; m += THREADS)
    bufA[m] = cmul(bufB[m & 255], bufB[256 + (m >> 8)]);
  __syncthreads();

  float2* cur = bufA;
  float2* alt = bufB;

  const int cl = lane & 15;     // pair slot within tile (N column)
  const int hi = lane >> 4;     // 0: amp0 / K cols {0,1}; 1: amp1 / K cols {2,3}
  const int r4 = lane & 3;      // M4 row (A replicates M4 4x down the rows)

  for (int l = 0; l < NLAYERS; ++l) {
    if (l > 0) {
      // fused RX*RZ*RY gate per qubit for this layer
      if (tid < NQ) {
        float2 g00, g01, g10, g11;
        fuse_gate(w[(l * NQ + tid) * 3 + 0], w[(l * NQ + tid) * 3 + 1],
                  w[(l * NQ + tid) * 3 + 2], g00, g01, g10, g11);
        gm[tid * 4 + 0] = g00; gm[tid * 4 + 1] = g01;
        gm[tid * 4 + 2] = g10; gm[tid * 4 + 3] = g11;
      }
      __syncthreads();

      for (int q = 0; q < NQ; ++q) {
        // ---- A operand: real 4x4 M4 of the complex 2x2 gate, replicated over
        // the 16 rows. A[m][k]: lane = (m&15) + 16*(k>=2), VGPR = k&1.
        // M4 row pattern: real row (u.x, -u.y), imag row (u.y, u.x), u = g[r][col].
        float2 u = gm[q * 4 + (r4 >> 1) * 2 + hi];
        v2f aop;
        if (r4 & 1) { aop.x = u.y;  aop.y =  u.x; }
        else        { aop.x = u.x;  aop.y = -u.y; }

        // Insert-zero-bit-at-q identity: k0(p) = p + (p & ~(2^q - 1)).
        const int himask = ~((1 << q) - 1);
        const int hoff   = hi << q;

        int   p  = wid * 16 + cl;                 // pair index, step 128/tile
        int   e0 = p + (p & himask) + hoff;       // element index of this lane's amp
        float2 bv = cur[e0];                      // ds_load_b64, all 32 lanes

        v8f czero = {0.f, 0.f, 0.f, 0.f, 0.f, 0.f, 0.f, 0.f};

        // fixed trip count, last iteration peeled -> unconditional prefetch,
        // scalar induction variable, no exec-mask juggling
        #pragma unroll 2
        for (int it = 0; it < TPW - 1; ++it) {
          p += 16 * NWAVES;
          int   e1 = p + (p & himask) + hoff;
          float2 bn = cur[e1];                    // prefetch next tile
          v2f bop; bop.x = bv.x; bop.y = bv.y;    // B[k][n] = component k of pair n
          v8f d = __builtin_amdgcn_wmma_f32_16x16x4_f32(
              false, aop, false, bop, (short)0, czero, false, false);
          // lanes<16: comps (0,1)=amp0 in d[0],d[1]; lanes>=16: comps (2,3)=amp1
          // in rows 10,11 = d[2],d[3]  (D rows repeat M4 with period 4)
          float2 res = hi ? make_float2(d[2], d[3]) : make_float2(d[0], d[1]);
          cur[e0] = res;                          // ds_store_b64, all 32 lanes
          e0 = e1; bv = bn;
        }
        {  // epilogue tile
          v2f bop; bop.x = bv.x; bop.y = bv.y;
          v8f d = __builtin_amdgcn_wmma_f32_16x16x4_f32(
              false, aop, false, bop, (short)0, czero, false, false);
          float2 res = hi ? make_float2(d[2], d[3]) : make_float2(d[0], d[1]);
          cur[e0] = res;
        }
        __syncthreads();
      }
    }

    // ---- CX ladder CX(0,1)..CX(12,13) == single permutation:
    // dst[m] = src[(m ^ (m<<1)) & mask]  (inverse prefix-XOR of bits)
    for (int m = tid; m < NSTATES; m += THREADS)
      alt[m] = cur[(m ^ (m << 1)) & (NSTATES - 1)];
    __syncthreads();
    float2* tswap = cur; cur = alt; alt = tswap;
  }
  // 4 swaps -> cur == bufA holds the final state

  // ---- <Z_j> epilogue.  m = tid + 256*i:
  //   bits 0-4 = lane (per-lane constant sign), bits 5-7 = wave id (per-wave
  //   constant), bits 8-13 = i (compile-time constant after full unroll).
  float psum = 0.f;
  float acch[6];
  #pragma unroll
  for (int jj = 0; jj < 6; ++jj) acch[jj] = 0.f;
  #pragma unroll
  for (int i = 0; i < NSTATES / THREADS; ++i) {   // 64 iters, signs fold to add/sub
    float2 a2 = cur[tid + (i << 8)];
    float p = __fmaf_rn(a2.x, a2.x, a2.y * a2.y);
    psum += p;
    #pragma unroll
    for (int jj = 0; jj < 6; ++jj)
      acch[jj] += ((i >> jj) & 1) ? -p : p;
  }
  float accl[5];
  #pragma unroll
  for (int j = 0; j < 5; ++j)
    accl[j] = ((lane >> j) & 1) ? -psum : psum;    // lane-bit signs before reduce
  #pragma unroll
  for (int off = 16; off > 0; off >>= 1) {
    psum += __shfl_xor(psum, off, 32);
    #pragma unroll
    for (int j = 0; j < 5; ++j)  accl[j] += __shfl_xor(accl[j], off, 32);
    #pragma unroll
    for (int jj = 0; jj < 6; ++jj) acch[jj] += __shfl_xor(acch[jj], off, 32);
  }
  if (lane == 0) {
    #pragma unroll
    for (int j = 0; j < 5; ++j) atomicAdd(&red[j], accl[j]);          // ds_add_f32
    #pragma unroll
    for (int j = 5; j < 8; ++j)
      atomicAdd(&red[j], ((wid >> (j - 5)) & 1) ? -psum : psum);      // wave-bit signs
    #pragma unroll
    for (int jj = 0; jj < 6; ++jj) atomicAdd(&red[8 + jj], acch[jj]);
  }
  __syncthreads();
  // output column i corresponds to qubit bit (13 - i)
  if (tid < NQ) out[b * NQ + tid] = red[NQ - 1 - tid];
}

extern "C" void kernel_launch(void* const* d_in, const int* in_sizes, int n_in,
                              void* d_out, int out_size, void* d_ws, size_t ws_size,
                              hipStream_t stream) {
  (void)n_in; (void)d_ws; (void)ws_size; (void)out_size;
  const float* x = (const float*)d_in[0];   // (B, 14) float32
  const float* w = (const float*)d_in[1];   // (4, 14, 3) float32
  float* out = (float*)d_out;               // (B, 14) float32
  int B = in_sizes[0] / NQ;                 // 2048
  // 2 x 128KB state buffers + small gate/reduction area, within 320KB WGP LDS
  size_t shmem = (size_t)(2 * NSTATES) * sizeof(float2) + 4096;
  qsim14_kernel<<<dim3(B), dim3(THREADS), shmem, stream>>>(x, w, out);
}